// TemporalExtensionShift_63814624084234
// MI455X (gfx1250) — compile-verified
//
#include <hip/hip_runtime.h>
#include <hip/hip_bf16.h>

// Temporal shift (depthwise dilated conv, one-hot taps) as a TDM-driven plane
// copy on gfx1250: each (n,c,t) output plane is either a pure 12.5KB plane
// copy (tensor_load_to_lds -> s_wait_tensorcnt -> tensor_store_from_lds),
// a zero plane, or (general weights) a 3-tap FMA fallback.
// Traffic: 411MB in + 411MB out -> ~35us roofline @ 23.3 TB/s.

typedef float        v4f __attribute__((ext_vector_type(4)));
typedef unsigned int v4u __attribute__((ext_vector_type(4)));
typedef int          v4i __attribute__((ext_vector_type(4)));
typedef int          v8i __attribute__((ext_vector_type(8)));

#define T_DIM 16
#define C_DIM 256
#define HW    3136          // 56*56 floats per (n,c,t) plane (contiguous)
#define HW4   (HW / 4)      // 784 float4 per plane
#define BLK   32            // one wave32 per block: one TDM descriptor pair each

#if __has_builtin(__builtin_amdgcn_tensor_load_to_lds) && \
    __has_builtin(__builtin_amdgcn_tensor_store_from_lds) && \
    __has_builtin(__builtin_amdgcn_s_wait_tensorcnt)
#define USE_TDM 1
#else
#define USE_TDM 0
#endif

__device__ __forceinline__ float uniform_f32(float v) {
    return __builtin_bit_cast(float,
        __builtin_amdgcn_readfirstlane(__builtin_bit_cast(int, v)));
}

#if USE_TDM
// D# group 0: {count=1(valid), lds_addr, global_addr[56:0], type=2("image")}
__device__ __forceinline__ v4u tdm_g0(unsigned lds_addr, unsigned long long ga) {
    v4u g;
    g.x = 0x1u;                                            // count=1, is_restore=0
    g.y = lds_addr;                                        // LDS byte address
    g.z = (unsigned)(ga & 0xFFFFFFFFu);                    // global_addr[31:0]
    g.w = (unsigned)((ga >> 32) & 0x01FFFFFFu) | (2u << 30); // addr[56:32] | type=2
    return g;
}

// D# group 1: 1-D tile of `nelem` 4-byte elements, no padding/iterate/multicast.
__device__ __forceinline__ v8i tdm_g1(unsigned nelem) {
    v8i g;
    g[0] = (int)(2u << 16);                                  // data_size=2 -> 4B
    g[1] = (int)((nelem & 0xFFFFu) << 16);                   // tensor_dim0[15:0]  @ 63:48
    g[2] = (int)(((nelem >> 16) & 0xFFFFu) | (1u << 16));    // tensor_dim0[31:16], tensor_dim1=1
    g[3] = (int)((nelem & 0xFFFFu) << 16);                   // tile_dim0 @ 127:112
    g[4] = 1;                                                // tile_dim1=1, tile_dim2=0
    g[5] = (int)nelem;                                       // tensor_dim0_stride[31:0]
    g[6] = 0;                                                // stride0[47:32], stride1[15:0]
    g[7] = 0;                                                // stride1[47:16]
    return g;
}

__device__ __forceinline__ void tdm_plane_copy(const float* src, float* dst,
                                               unsigned lds_addr) {
    const v4i z4 = {0, 0, 0, 0};                // groups 2/3 unused (<=2D tensor)
    const v8i z8 = {0, 0, 0, 0, 0, 0, 0, 0};    // 5th arg (6-arg toolchain form)
    const v8i g1 = tdm_g1(HW);
    __builtin_amdgcn_tensor_load_to_lds(
        tdm_g0(lds_addr, (unsigned long long)(uintptr_t)src), g1, z4, z4, z8, 0);
    __builtin_amdgcn_s_wait_tensorcnt(0);       // LDS RAW: load landed before store
    __builtin_amdgcn_tensor_store_from_lds(
        tdm_g0(lds_addr, (unsigned long long)(uintptr_t)dst), g1, z4, z4, z8, 0);
}
#endif

__global__ __launch_bounds__(BLK)
void tshift_kernel(const float* __restrict__ x,
                   const float* __restrict__ w,   // [C][1][3] flattened
                   float* __restrict__ y)
{
    __shared__ float stage[HW];                  // 12544 B staging plane

    const int p = blockIdx.x;                    // plane id: ((n*C + c)*T + t)
    const int t = p & (T_DIM - 1);
    const int c = (p >> 4) & (C_DIM - 1);

    // Per-channel taps, wave-uniform: y[t] = w0*x[t-2] + w1*x[t] + w2*x[t+2]
    const float w0 = uniform_f32(w[3 * c + 0]);
    const float w1 = uniform_f32(w[3 * c + 1]);
    const float w2 = uniform_f32(w[3 * c + 2]);

    const int  nz   = (w0 != 0.0f) + (w1 != 0.0f) + (w2 != 0.0f);
    const float wv  = (w0 != 0.0f) ? w0 : (w1 != 0.0f) ? w1 : w2;
    const int  dt   = (w0 != 0.0f) ? -2 : (w1 != 0.0f) ? 0 : 2;
    const bool pure = (nz == 1) && (wv == 1.0f); // plane copy (reference weights)

    const long base = (long)p * HW;              // plane base in floats
    v4f* __restrict__ yv = (v4f*)(y + base);

    if (pure || nz == 0) {
        const int ts = t + dt;
        if (nz != 0 && ts >= 0 && ts < T_DIM) {
#if USE_TDM
            // Pure shift: DMA plane copy, no VGPR data movement.
            const unsigned lds_addr = (unsigned)(uintptr_t)stage; // LDS offset = addr[31:0]
            tdm_plane_copy(x + base + (long)dt * HW, y + base, lds_addr);
#else
            const v4f* __restrict__ xv = (const v4f*)(x + base + (long)dt * HW);
            for (int i = threadIdx.x; i < HW4; i += BLK)
                __builtin_nontemporal_store(__builtin_nontemporal_load(xv + i), yv + i);
#endif
        } else {
            // Out-of-range source (or all-zero taps): zero plane.
            const v4f zero = (v4f)0.0f;
            for (int i = threadIdx.x; i < HW4; i += BLK)
                __builtin_nontemporal_store(zero, yv + i);
        }
    } else {
        // General weights: 3-tap dilated depthwise conv (wave-uniform taps).
        const bool tap0 = (w0 != 0.0f) && (t >= 2);
        const bool tap1 = (w1 != 0.0f);
        const bool tap2 = (w2 != 0.0f) && (t < T_DIM - 2);
        const v4f* __restrict__ xv = (const v4f*)x;
        const long b4 = (long)p * HW4;
        for (int i = threadIdx.x; i < HW4; i += BLK) {
            v4f acc = (v4f)0.0f;
            if (tap0) acc += w0 * __builtin_nontemporal_load(xv + (b4 - 2 * HW4) + i);
            if (tap1) acc += w1 * __builtin_nontemporal_load(xv + b4 + i);
            if (tap2) acc += w2 * __builtin_nontemporal_load(xv + (b4 + 2 * HW4) + i);
            __builtin_nontemporal_store(acc, yv + i);
        }
    }
}

extern "C" void kernel_launch(void* const* d_in, const int* in_sizes, int n_in,
                              void* d_out, int out_size, void* d_ws, size_t ws_size,
                              hipStream_t stream)
{
    const float* x = (const float*)d_in[0];   // [8,256,16,56,56] f32
    const float* w = (const float*)d_in[1];   // [256,1,3] f32
    float*       y = (float*)d_out;

    const int nplanes = in_sizes[0] / HW;     // 8*256*16 = 32768
    tshift_kernel<<<nplanes, BLK, 0, stream>>>(x, w, y);
}